// DGL_RGCN_RL_35974646071549
// MI455X (gfx1250) — compile-verified
//
#include <hip/hip_runtime.h>
#include <hip/hip_bf16.h>

// RGCN (basis decomposition) on MI455X / gfx1250.
// Per layer: compose Wall = [W_r0..W_r5 | S] in a K-pair-packed float2 layout,
// run a wave32 WMMA f32 16x16x4 GEMM (each wave computes NT 16x16 col tiles,
// reusing the A fragment), then coalesced gather + atomic scatter over edges,
// then ReLU. FP32 throughout (workload is HBM/atomic bound; no precision loss).

typedef __attribute__((ext_vector_type(2))) float v2f;
typedef __attribute__((ext_vector_type(8))) float v8f;

#define NRELS 6
#define NBASES 6

// ---------------------------------------------------------------- pad layer-1 input
__global__ void rgcn_pad(const float* __restrict__ x, float* __restrict__ hp,
                         int N, int inDim, int padDim) {
    long long idx = (long long)blockIdx.x * blockDim.x + threadIdx.x;
    long long total = (long long)N * padDim;
    if (idx >= total) return;
    int n = (int)(idx / padDim);
    int c = (int)(idx % padDim);
    hp[idx] = (c < inDim) ? x[(long long)n * inDim + c] : 0.0f;
}

// ---------------------------------------------------------------- build Wall (packed)
// Logical Wall[k][r*out + o]; stored K-pair packed:
//   Wpk[(k>>1)*wc + n] is a float2 holding rows {2j, 2j+1} at column n.
// Flat float index: ((k>>1)*wc + n)*2 + (k&1).
__global__ void rgcn_wall(const float* __restrict__ V, const float* __restrict__ C,
                          const float* __restrict__ S,
                          int inDim, int Kpad, int outDim, float* __restrict__ Wpk) {
    int wc = (NRELS + 1) * outDim;
    long long idx = (long long)blockIdx.x * blockDim.x + threadIdx.x;
    long long total = (long long)Kpad * wc;
    if (idx >= total) return;
    int k = (int)(idx / wc);
    int cidx = (int)(idx % wc);
    int r = cidx / outDim;
    int o = cidx % outDim;
    float val = 0.0f;
    if (k < inDim) {
        if (r == NRELS) {
            val = S[(long long)k * outDim + o];
        } else {
            #pragma unroll
            for (int b = 0; b < NBASES; ++b)
                val += C[r * NBASES + b] * V[((long long)b * inDim + k) * outDim + o];
        }
    }
    Wpk[((long long)(k >> 1) * wc + cidx) * 2 + (k & 1)] = val;
}

// ---------------------------------------------------------------- WMMA GEMM
// One wave computes a 16 x (16*NT) output strip. A fragment loaded once per
// K-step (b64), B fragments loaded as single b64 each (K-pair packed layout),
// NT wmma f32 16x16x4 per K-step. Pointers bumped by constant strides.
//
// f32 16x16x4 VGPR layout (ISA 7.12.2):
//   A: lane l -> row M = l&15, K pair kb = 2*(l>>4): a = {A[M][kb], A[M][kb+1]}
//   B: lane l -> col N = l&15, same K pair:          b = {B[kb][N], B[kb+1][N]}
//   C/D: VGPR v -> M = v + 8*(l>>4), N = l&15
template <int NT>
__global__ __launch_bounds__(32) void rgcn_gemm(
        const float* __restrict__ A, int lda,
        const float* __restrict__ Wpk, int wc, int K,
        float* __restrict__ tmp, int outDim,
        float* __restrict__ selfOut, const float* __restrict__ bias) {
    const int l    = threadIdx.x;
    const int row0 = blockIdx.x * 16;
    const int col0 = blockIdx.y * (16 * NT);
    const int m    = l & 15;
    const int half = l >> 4;                 // 0 or 1
    const int jb   = half;                   // K-pair index offset (kb = 2*half)

    const v2f* __restrict__ ap =
        (const v2f*)(A + (size_t)(row0 + m) * lda) + jb;
    const v2f* __restrict__ bp =
        (const v2f*)Wpk + (size_t)jb * wc + (col0 + m);

    v8f c[NT];
    #pragma unroll
    for (int t = 0; t < NT; ++t) c[t] = (v8f){};

    for (int k = 0; k < K; k += 4) {
        v2f a = *ap;
        ap += 2;                             // advance 4 floats
        #pragma unroll
        for (int t = 0; t < NT; ++t) {
            v2f b = bp[16 * t];
            // 8 args: (neg_a, A, neg_b, B, c_mod, C, reuse_a, reuse_b)
            c[t] = __builtin_amdgcn_wmma_f32_16x16x4_f32(
                       false, a, false, b, (short)0, c[t], false, false);
        }
        bp += (size_t)2 * wc;                // advance 2 K-pair rows
    }

    const int selfBase = wc - outDim;        // 6*out (always a 16*NT multiple)
    const int mOff = half << 3;              // 0 or 8
    #pragma unroll
    for (int t = 0; t < NT; ++t) {
        const int n = col0 + 16 * t + m;
        if (n >= selfBase) {
            const int oc = n - selfBase;
            const float bv = bias[oc];
            #pragma unroll
            for (int v = 0; v < 8; ++v)
                selfOut[(size_t)(row0 + v + mOff) * outDim + oc] = c[t][v] + bv;
        } else {
            #pragma unroll
            for (int v = 0; v < 8; ++v)
                tmp[(size_t)(row0 + v + mOff) * selfBase + n] = c[t][v];
        }
    }
}

// ---------------------------------------------------------------- edge scatter
// One thread per (edge, out-channel); outDim is a power of two (32/64/128) so
// lanes of a wave share an edge -> coalesced gather + coalesced atomic bursts.
__global__ void rgcn_scatter(const float* __restrict__ tmp, int tstride,
                             int outMask, int outShift,
                             const int* __restrict__ src, const int* __restrict__ dst,
                             const int* __restrict__ et,
                             float* __restrict__ agg, long long total) {
    long long idx = (long long)blockIdx.x * blockDim.x + threadIdx.x;
    if (idx >= total) return;
    int e = (int)(idx >> outShift);
    int c = (int)(idx & outMask);
    int s = src[e];
    int d = dst[e];
    int r = et[e];
    float v = tmp[(size_t)s * tstride + ((size_t)r << outShift) + c];
    atomicAdd(&agg[((size_t)d << outShift) + c], v);
}

// ---------------------------------------------------------------- ReLU (vec4)
__global__ void rgcn_relu4(float4* __restrict__ h, long long n4) {
    long long idx = (long long)blockIdx.x * blockDim.x + threadIdx.x;
    if (idx >= n4) return;
    float4 v = h[idx];
    v.x = fmaxf(v.x, 0.0f); v.y = fmaxf(v.y, 0.0f);
    v.z = fmaxf(v.z, 0.0f); v.w = fmaxf(v.w, 0.0f);
    h[idx] = v;
}

// ---------------------------------------------------------------- driver
static inline int cdiv(long long a, long long b) { return (int)((a + b - 1) / b); }
static inline int ilog2(int x) { int s = 0; while ((1 << s) < x) ++s; return s; }

extern "C" void kernel_launch(void* const* d_in, const int* in_sizes, int n_in,
                              void* d_out, int out_size, void* d_ws, size_t ws_size,
                              hipStream_t stream) {
    const float* node_feats = (const float*)d_in[0];
    const int*   src        = (const int*)d_in[1];
    const int*   dst        = (const int*)d_in[2];
    const int*   etypes     = (const int*)d_in[3];
    const float* V[3]  = { (const float*)d_in[4],  (const float*)d_in[8],  (const float*)d_in[12] };
    const float* C[3]  = { (const float*)d_in[5],  (const float*)d_in[9],  (const float*)d_in[13] };
    const float* S[3]  = { (const float*)d_in[6],  (const float*)d_in[10], (const float*)d_in[14] };
    const float* bb[3] = { (const float*)d_in[7],  (const float*)d_in[11], (const float*)d_in[15] };

    const int dims[4] = { 26, 64, 128, 32 };
    const int N = in_sizes[0] / dims[0];     // 100000 (multiple of 16)
    const int E = in_sizes[1];               // 600000

    // workspace carve-up (floats); total ~397 MB
    float* ws   = (float*)d_ws;
    float* hpad = ws;                                  // N * 32
    float* h1   = hpad + (size_t)N * 32;               // N * 64
    float* h2   = h1   + (size_t)N * 64;               // N * 128
    float* tmp  = h2   + (size_t)N * 128;              // N * 6*128 (max, reused)
    float* wall = tmp  + (size_t)N * 6 * 128;          // 128 * 7*128 (max, reused)

    const int TB = 256;

    // pad layer-1 input: [N,26] -> [N,32] (zero tail)
    rgcn_pad<<<cdiv((long long)N * 32, TB), TB, 0, stream>>>(node_feats, hpad, N, dims[0], 32);

    const float* Aptr[3] = { hpad, h1, h2 };
    const int    lda[3]  = { 32, 64, 128 };
    const int    Kd[3]   = { 28, 64, 128 };            // K padded to mult. of 4
    float* outH[3] = { h1, h2, (float*)d_out };

    for (int L = 0; L < 3; ++L) {
        const int inD  = dims[L];
        const int outD = dims[L + 1];
        const int wc   = (NRELS + 1) * outD;           // 448 / 896 / 224
        const int ts   = NRELS * outD;                 // tmp row stride
        const int sh   = ilog2(outD);

        // 1) compose Wall in K-pair-packed layout (padded rows zeroed)
        rgcn_wall<<<cdiv((long long)Kd[L] * wc, TB), TB, 0, stream>>>(
            V[L], C[L], S[L], inD, Kd[L], outD, wall);

        // 2) WMMA GEMM: tmp = h @ [W_r], outH = h @ S + b
        if (wc % 64 == 0) {
            dim3 g(N / 16, wc / 64);
            rgcn_gemm<4><<<g, 32, 0, stream>>>(
                Aptr[L], lda[L], wall, wc, Kd[L], tmp, outD, outH[L], bb[L]);
        } else {
            dim3 g(N / 16, wc / 32);
            rgcn_gemm<2><<<g, 32, 0, stream>>>(
                Aptr[L], lda[L], wall, wc, Kd[L], tmp, outD, outH[L], bb[L]);
        }

        // 3) edge gather + scatter-add into outH
        long long etotal = (long long)E << sh;
        rgcn_scatter<<<cdiv(etotal, TB), TB, 0, stream>>>(
            tmp, ts, outD - 1, sh, src, dst, etypes, outH[L], etotal);

        // 4) ReLU
        rgcn_relu4<<<cdiv(((long long)N * outD) / 4, TB), TB, 0, stream>>>(
            (float4*)outH[L], ((long long)N * outD) / 4);
    }
}